// DamerauLevenshtein_68616397521353
// MI455X (gfx1250) — compile-verified
//
#include <hip/hip_runtime.h>

#define BSZ        2
#define SEQ        16
#define NUM_WORDS  10000
#define MAXLEN     14
#define NUM_CHARS  30
#define L2         (MAXLEN + 2)   // 16
#define NT         64             // threads per block (2 wave32s)

__device__ __forceinline__ int imin(int a, int b) { return a < b ? a : b; }

__global__ __launch_bounds__(NT)
void DamerauLevenshtein_68616397521353_kernel(const int* __restrict__ x,
                                              const int* __restrict__ words,
                                              const int* __restrict__ wlens,
                                              float* __restrict__ out) {
    // DP table, column-per-thread layout: dmat[cell][tid] (u16, 32 KB).
    __shared__ unsigned short dmat[L2 * L2][NT];
    // This block's word chunk, staged via async global->LDS copies (3.5 KB).
    __shared__ int wbuf[NT * MAXLEN];
    __shared__ int xrow[MAXLEN + 1];

    const int bs = blockIdx.y;                 // b*SEQ + s, 0..31
    const int t  = threadIdx.x;
    int w        = blockIdx.x * NT + t;
    const bool active = (w < NUM_WORDS);
    if (!active) w = NUM_WORDS - 1;

    // ---- coalesced async staging of 64 words x 14 dwords into LDS ----
    // Low 32 bits of a generic pointer to __shared__ == LDS byte offset.
    {
        const unsigned lds_base = (unsigned)(unsigned long long)(void*)&wbuf[0];
        const int chunk0 = blockIdx.x * (NT * MAXLEN);
#pragma unroll
        for (int r = 0; r < MAXLEN; ++r) {
            int idx = imin(chunk0 + r * NT + t, NUM_WORDS * MAXLEN - 1); // in-bounds clamp
            unsigned long long gaddr = (unsigned long long)(words + idx);
            unsigned laddr = lds_base + (unsigned)(r * NT + t) * 4u;
            asm volatile("global_load_async_to_lds_b32 %0, %1, off"
                         :: "v"(laddr), "v"(gaddr) : "memory");
        }
    }

    // Overlap with async: stage the shared query row.
    if (t < MAXLEN + 1) xrow[t] = x[bs * (MAXLEN + 1) + t];
    const int wlen = wlens[w];                 // coalesced

    asm volatile("s_wait_asynccnt 0" ::: "memory");   // drain this wave's async copies
    __syncthreads();                                  // make wbuf/xrow block-visible
    if (!active) return;                              // no barriers follow

    // swl = argmax over x[b,s,:] (first occurrence of max)
    int swl = 0, best = xrow[0];
#pragma unroll
    for (int c = 1; c <= MAXLEN; ++c) {
        int v = xrow[c];
        if (v > best) { best = v; swl = c; }
    }

    // This thread's word chars from LDS (bank-conflict-free: 14t mod 64 injective).
    int wc[MAXLEN];
#pragma unroll
    for (int j = 0; j < MAXLEN; ++j) wc[j] = wbuf[t * MAXLEN + j];

    const int maxd = swl + wlen;

    // ---- init: only row 0, row 1, col 0, col 1 are ever read before the DP
    //      writes them (gathers reach only completed rows k <= i-1). ----
#pragma unroll
    for (int c = 0; c < L2; ++c) {
        dmat[0 * L2 + c][t] = (unsigned short)maxd;                                // row 0
        dmat[1 * L2 + c][t] =
            (unsigned short)((c == 0) ? maxd : (((c - 1) <= wlen) ? (c - 1) : 0)); // row 1 (wla)
    }
#pragma unroll
    for (int r = 2; r < L2; ++r) {
        dmat[r * L2 + 0][t] = (unsigned short)maxd;                                // col 0
        dmat[r * L2 + 1][t] = (unsigned short)(((r - 1) <= swl) ? (r - 1) : 0);    // col 1 (swl_row)
    }

    // prev = row i (starts as row 1), register resident.
    int prev[L2];
#pragma unroll
    for (int c = 0; c < L2; ++c)
        prev[c] = (c == 0) ? maxd : (((c - 1) <= wlen) ? (c - 1) : 0);

    // kreg[j] == da[wc[j-1]] in registers: last row i' with xc_{i'} == wc[j-1].
    int kreg[MAXLEN + 1];
#pragma unroll
    for (int j = 1; j <= MAXLEN; ++j) kreg[j] = 0;

    for (int i = 1; i <= MAXLEN; ++i) {
        const int xc = xrow[i - 1];            // uniform LDS broadcast read

        // db / l chain: fully determined before the row starts (pure VALU)
        int ll[MAXLEN + 1];
        {
            int db = 0;
#pragma unroll
            for (int j = 1; j <= MAXLEN; ++j) {
                ll[j] = db;
                if (xc == wc[j - 1]) db = j;
            }
        }

        // Batched transposition gathers: 14 independent ds_loads, one wait.
        int dtr[MAXLEN + 1];
#pragma unroll
        for (int j = 1; j <= MAXLEN; ++j)
            dtr[j] = (int)dmat[kreg[j] * L2 + ll[j]][t];

        // Arithmetic chain + persist row i+1 for future gathers.
        int cur[L2];
        cur[0] = maxd;
        cur[1] = (i <= swl) ? i : 0;
#pragma unroll
        for (int j = 1; j <= MAXLEN; ++j) {
            const int cost = (xc != wc[j - 1]) ? 1 : 0;
            const int c4   = dtr[j] + (i - kreg[j]) + (j - ll[j]) - 1;  // transposition
            const int v    = imin(imin(prev[j + 1] + 1, cur[j] + 1),
                                  imin(prev[j] + cost, c4));
            cur[j + 1] = v;
            dmat[(i + 1) * L2 + (j + 1)][t] = (unsigned short)v;
        }

#pragma unroll
        for (int c = 0; c < L2; ++c) prev[c] = cur[c];

        // da[xc] = i  ->  register update, no LDS, no barriers.
#pragma unroll
        for (int j = 1; j <= MAXLEN; ++j)
            if (xc == wc[j - 1]) kreg[j] = i;
    }

    const int res = (int)dmat[(swl + 1) * L2 + (wlen + 1)][t];
    out[bs * NUM_WORDS + w] = (float)res;
}

extern "C" void kernel_launch(void* const* d_in, const int* in_sizes, int n_in,
                              void* d_out, int out_size, void* d_ws, size_t ws_size,
                              hipStream_t stream) {
    (void)in_sizes; (void)n_in; (void)out_size; (void)d_ws; (void)ws_size;
    const int* x     = (const int*)d_in[0];
    const int* words = (const int*)d_in[1];
    const int* wlens = (const int*)d_in[2];
    // d_in[3] = num_chars scalar; compile-time constant here.
    float* out = (float*)d_out;

    dim3 grid((NUM_WORDS + NT - 1) / NT, BSZ * SEQ);
    dim3 block(NT);
    hipLaunchKernelGGL(DamerauLevenshtein_68616397521353_kernel,
                       grid, block, 0, stream, x, words, wlens, out);
}